// LDStackInner_56822417326796
// MI455X (gfx1250) — compile-verified
//
#include <hip/hip_runtime.h>
#include <math.h>

#define T_LEN 1024
#define B_SZ  32
#define K_SZ  8
#define N_SZ  64
#define M_SZ  64
#define TT    16
#define NTILES (T_LEN / TT)

typedef __attribute__((ext_vector_type(2))) float v2f;
typedef __attribute__((ext_vector_type(8))) float v8f;

// ---------------------------------------------------------------------------
// Kernel 1: Bp[k,j] = exp(-sum_{i!=j} log(1 - lam[k,i]/lam[k,j]))  (complex)
// ---------------------------------------------------------------------------
__global__ void compute_bp_kernel(const float* __restrict__ lam_re,
                                  const float* __restrict__ lam_im,
                                  float* __restrict__ bp_re,
                                  float* __restrict__ bp_im) {
    int tid = threadIdx.x;              // 0..511  -> (k, j)
    int k = tid >> 6, j = tid & 63;
    float cr = lam_re[k * N_SZ + j], ci = lam_im[k * N_SZ + j];
    float inv = 1.0f / (cr * cr + ci * ci);
    float sre = 0.0f, sim = 0.0f;       // accumulates -sum log(1 - ratio)
    for (int i = 0; i < N_SZ; ++i) {
        if (i == j) continue;
        float ar = lam_re[k * N_SZ + i], ai = lam_im[k * N_SZ + i];
        // ratio = lam_i / lam_j
        float rr = (ar * cr + ai * ci) * inv;
        float ri = (ai * cr - ar * ci) * inv;
        float zr = 1.0f - rr, zi = -ri;
        sre -= 0.5f * logf(zr * zr + zi * zi);
        sim -= atan2f(zi, zr);
    }
    float e = expf(sre);
    bp_re[tid] = e * cosf(sim);
    bp_im[tid] = e * sinf(sim);
}

// ---------------------------------------------------------------------------
// Kernel 2: fused 3-pass lockstep recurrence + WMMA projection + k-mean
// Grid = B blocks, 1024 threads (32 waves). Threads <512 own (k,n) states.
// Wave w handles WMMA job (k = w&7, mtile = w>>3).
// ---------------------------------------------------------------------------
__global__ __launch_bounds__(1024) void ssm_fused_kernel(
    const float* __restrict__ x,
    const float* __restrict__ lam_re, const float* __restrict__ lam_im,
    const float* __restrict__ C_re,   const float* __restrict__ C_im,
    const float* __restrict__ Dmat,   const float* __restrict__ Do,
    const float* __restrict__ bp_re,  const float* __restrict__ bp_im,
    float* __restrict__ out) {
    extern __shared__ float smem[];
    float* s3re  = smem;                          // TT*K*N = 8192 floats
    float* s3im  = s3re + TT * K_SZ * N_SZ;       // 8192
    float* stage = s3im + TT * K_SZ * N_SZ;       // TT*M*2 = 2048 (re,im interleaved)
    float* xs    = stage + TT * M_SZ * 2;         // TT*K = 128
    float* Dl    = xs + TT * K_SZ;                // K*M = 512
    float* sDo   = Dl + K_SZ * M_SZ;              // M = 64

    const int tid  = threadIdx.x;
    const int b    = blockIdx.x;
    const int wave = tid >> 5;
    const int lane = tid & 31;

    // ---- one-time LDS init ------------------------------------------------
    if (tid < K_SZ * M_SZ) Dl[tid] = Dmat[tid];
    if (tid < M_SZ) {
        float s = 0.0f;
        for (int k = 0; k < K_SZ; ++k) s += Do[k * M_SZ + tid];
        sDo[tid] = s;
    }
    stage[tid] = 0.0f;
    stage[tid + 1024] = 0.0f;

    // ---- recurrence-owner state (threads < 512 own (rk, rn)) -------------
    const int rk = tid >> 6;
    float lre = 0.f, lim = 0.f, bre = 0.f, bim = 0.f;
    float s1r = 0.f, s1i = 0.f, s2r = 0.f, s2i = 0.f, s3r = 0.f, s3i = 0.f;
    if (tid < 512) {
        lre = lam_re[tid]; lim = lam_im[tid];   // flat (K,N) index == tid
        bre = bp_re[tid];  bim = bp_im[tid];
    }

    // ---- WMMA job: preload loop-invariant B operands (C slices) ----------
    // B is 4x16 (K x N): VGPR0 = rows {0 | 2} (lanes<16 | >=16), VGPR1 = rows {1 | 3}
    const int jk    = wave & 7;
    const int mtile = wave >> 3;
    const int colm  = mtile * 16 + (lane & 15);
    const int khalf = lane >> 4;

    v2f Bre[16], Bim[16];
#pragma unroll
    for (int kk = 0; kk < 16; ++kk) {
        int n0 = 4 * kk + 2 * khalf;
        const float* pr = C_re + (size_t)(jk * N_SZ + n0) * M_SZ + colm;
        const float* pi = C_im + (size_t)(jk * N_SZ + n0) * M_SZ + colm;
        v2f br; br.x = pr[0]; br.y = pr[M_SZ];
        v2f bi; bi.x = pi[0]; bi.y = pi[M_SZ];
        Bre[kk] = br; Bim[kk] = bi;
    }

    __syncthreads();

    for (int tile = 0; tile < NTILES; ++tile) {
        // ---- stage x[t,b,:] for writeout phase ---------------------------
        if (tid < TT * K_SZ) {
            int tt = tid >> 3, k = tid & 7;
            xs[tid] = x[(size_t)((tile * TT + tt) * B_SZ + b) * K_SZ + k];
        }
        // ---- 16 lockstep recurrence steps (3 passes fused) ---------------
        if (tid < 512) {
            for (int tt = 0; tt < TT; ++tt) {
                int t = tile * TT + tt;
                if (t > 0) {
                    float xv = x[(size_t)((t - 1) * B_SZ + b) * K_SZ + rk];
                    float bxr = bre * xv, bxi = bim * xv;
                    float l1r = lre * s1r - lim * s1i, l1i = lre * s1i + lim * s1r;
                    float l2r = lre * s2r - lim * s2i, l2i = lre * s2i + lim * s2r;
                    float l3r = lre * s3r - lim * s3i, l3i = lre * s3i + lim * s3r;
                    float a1 = rsqrtf(1.0f + l1r * l1r + l1i * l1i); // alpha from pass-1 state
                    float a2 = rsqrtf(1.0f + l2r * l2r + l2i * l2i); // alpha from pass-2 state
                    s1r = l1r + bxr;      s1i = l1i + bxi;
                    s2r = a1 * l2r + bxr; s2i = a1 * l2i + bxi;
                    s3r = a2 * l3r + bxr; s3i = a2 * l3i + bxi;
                }
                s3re[(tt * K_SZ + rk) * N_SZ + (tid & 63)] = s3r;
                s3im[(tt * K_SZ + rk) * N_SZ + (tid & 63)] = s3i;
            }
        }
        __syncthreads();  // s3 tile + xs visible; stage is zeroed

        // ---- WMMA projection: Cs[t,m] += s3 . C  (complex, f32) ----------
        {
            v8f accre = {}; v8f accim = {};
            const int arow = lane & 15;  // A: lanes 0-15 / 16-31 both map to M rows 0-15
#pragma unroll
            for (int kk = 0; kk < 16; ++kk) {
                int n0 = 4 * kk + 2 * khalf;
                const v2f are = *(const v2f*)(s3re + (arow * K_SZ + jk) * N_SZ + n0);
                const v2f aim = *(const v2f*)(s3im + (arow * K_SZ + jk) * N_SZ + n0);
                v2f aimn = -aim;  // f32 WMMA has no A/B NEG modifier
                accre = __builtin_amdgcn_wmma_f32_16x16x4_f32(false, are,  false, Bre[kk], (short)0, accre, false, false);
                accre = __builtin_amdgcn_wmma_f32_16x16x4_f32(false, aimn, false, Bim[kk], (short)0, accre, false, false);
                accim = __builtin_amdgcn_wmma_f32_16x16x4_f32(false, are,  false, Bim[kk], (short)0, accim, false, false);
                accim = __builtin_amdgcn_wmma_f32_16x16x4_f32(false, aim,  false, Bre[kk], (short)0, accim, false, false);
            }
            // D layout: VGPR v -> row v (lanes<16) / v+8 (lanes>=16), col = lane&15.
            // Reduce the 8 k-waves of this mtile via LDS float atomics.
            const int rbase = khalf * 8;
            const int cold  = mtile * 16 + (lane & 15);
#pragma unroll
            for (int v = 0; v < 8; ++v) {
                int idx = ((rbase + v) * M_SZ + cold) * 2;
                atomicAdd(&stage[idx],     accre[v]);
                atomicAdd(&stage[idx + 1], accim[v]);
            }
        }
        __syncthreads();  // stage holds sum_k Cs

        // ---- writeout: y = (Cs + D*x + Do_sum)/K, interleaved complex ----
        {
            int tt = tid >> 6;
            int m  = tid & 63;
            int t  = tile * TT + tt;
            float csr = stage[(tt * M_SZ + m) * 2];
            float csi = stage[(tt * M_SZ + m) * 2 + 1];
            float dx = 0.0f;
#pragma unroll
            for (int k = 0; k < K_SZ; ++k) dx += xs[tt * K_SZ + k] * Dl[k * M_SZ + m];
            v2f o;
            o.x = (csr + dx + sDo[m]) * 0.125f;
            o.y = csi * 0.125f;
            *(v2f*)(out + ((size_t)(t * B_SZ + b) * M_SZ + m) * 2) = o;
        }
        __syncthreads();  // done reading stage
        stage[tid] = 0.0f;          // re-zero for next tile (visible after
        stage[tid + 1024] = 0.0f;   // the next barrier, before ds_adds)
    }
}

// ---------------------------------------------------------------------------
extern "C" void kernel_launch(void* const* d_in, const int* in_sizes, int n_in,
                              void* d_out, int out_size, void* d_ws, size_t ws_size,
                              hipStream_t stream) {
    const float* x      = (const float*)d_in[0];
    const float* lam_re = (const float*)d_in[1];
    const float* lam_im = (const float*)d_in[2];
    const float* C_re   = (const float*)d_in[3];
    const float* C_im   = (const float*)d_in[4];
    const float* Dmat   = (const float*)d_in[5];
    const float* Do     = (const float*)d_in[6];
    float* out   = (float*)d_out;              // (T,B,M) complex64 -> interleaved f32 pairs
    float* bp_re = (float*)d_ws;               // K*N
    float* bp_im = bp_re + K_SZ * N_SZ;        // K*N

    compute_bp_kernel<<<1, K_SZ * N_SZ, 0, stream>>>(lam_re, lam_im, bp_re, bp_im);

    size_t shmem = (size_t)(2 * TT * K_SZ * N_SZ + TT * M_SZ * 2 + TT * K_SZ +
                            K_SZ * M_SZ + M_SZ) * sizeof(float);
    ssm_fused_kernel<<<B_SZ, 1024, shmem, stream>>>(
        x, lam_re, lam_im, C_re, C_im, Dmat, Do, bp_re, bp_im, out);
}